// SteerablePyramid_53867479826824
// MI455X (gfx1250) — compile-verified
//
#include <hip/hip_runtime.h>
#include <math.h>

// ---------------------------------------------------------------------------
// Steerable pyramid on MI455X (gfx1250) as complex GEMMs on the fp32 WMMA
// matrix pipe (V_WMMA_F32_16X16X4_F32), with Tensor Data Mover (TDM) staging
// of unfiltered operand tiles into LDS (tensor_load_to_lds + TENSORcnt).
//
//   fft2(x)  = W @ x @ W^T              (W[u,m] = e^{-2pi i um/512})
//   band_j,k = R_f @ (Xf .* filt) @ R_f^T,  f = 2^(j+1)
//   R_f      = D_f @ IF   (downsample folded into the inverse DFT matrix)
// ---------------------------------------------------------------------------

typedef __attribute__((ext_vector_type(2))) float v2f;
typedef __attribute__((ext_vector_type(8))) float v8f;

#define WMMA_F32_16x16x4(a, b, c) \
  __builtin_amdgcn_wmma_f32_16x16x4_f32(false, (a), false, (b), (short)0, (c), false, false)

#define NFREQ 512
#define NIMG  12        // 4 batch * 3 channels
#define TPB   256       // 8 waves, wave32

// LDS tile: 64 rows x 16 complex (32 floats) + 1 pad dword per row -> stride 33.
#define TROW 33

// ---------------------------------------------------------------------------
// Tensor Data Mover: 2D tile load Global -> LDS (ISA ch.8 D# layout).
// This toolchain exposes the 6-arg builtin:
//   (uint32x4 g0, int32x8 g1, int32x4 g2, int32x4 g3, int32x8 extra, i32 cpol)
// ---------------------------------------------------------------------------
#if defined(__has_builtin)
#if __has_builtin(__builtin_amdgcn_tensor_load_to_lds)
#define HAVE_TDM 1
#endif
#endif
#ifndef HAVE_TDM
#define HAVE_TDM 0
#endif

#if HAVE_TDM
typedef unsigned int u32x4 __attribute__((ext_vector_type(4)));
typedef int          i32x4 __attribute__((ext_vector_type(4)));
typedef int          i32x8 __attribute__((ext_vector_type(8)));

// 2D tile: tileD1 rows x tileD0 4-byte elements, row stride strideD0 elements.
// tensorD0/tensorD1 = remaining tensor extent from the tile origin (OOB -> 0).
// Pad 1 dword after every 32 dwords stored -> LDS row stride 33 dwords.
__device__ __forceinline__ void tdm_load_tile_2d(unsigned ldsOff, const void* gptr,
                                                 unsigned tensorD0, unsigned tensorD1,
                                                 unsigned tileD0, unsigned tileD1,
                                                 unsigned strideD0) {
  unsigned long long ga = (unsigned long long)(uintptr_t)gptr;
  u32x4 g0;
  g0.x = 1u;                                               // count=1 (valid), user mode
  g0.y = ldsOff;                                           // LDS byte address
  g0.z = (unsigned)ga;                                     // global addr [31:0]
  g0.w = (unsigned)((ga >> 32) & 0x01FFFFFFu) | (2u << 30);// addr [56:32], type=2
  i32x8 g1;
  g1[0] = (int)((2u << 16) | (1u << 20) | (4u << 22));     // data_size=4B, pad_en, interval=32dw (amount=1dw)
  g1[1] = (int)((tensorD0 & 0xFFFFu) << 16);               // tensor_dim0[15:0] (barrier addr = 0)
  g1[2] = (int)((tensorD0 >> 16) | ((tensorD1 & 0xFFFFu) << 16));
  g1[3] = (int)((tensorD1 >> 16) | (tileD0 << 16));        // tile_dim0
  g1[4] = (int)tileD1;                                     // tile_dim1 (tile_dim2 = 0)
  g1[5] = (int)strideD0;                                   // tensor_dim0_stride [31:0]
  g1[6] = 0;
  g1[7] = 0;
  i32x4 z4 = {0, 0, 0, 0};
  i32x8 z8 = {0, 0, 0, 0, 0, 0, 0, 0};
  __builtin_amdgcn_tensor_load_to_lds(g0, g1, z4, z4, z8, 0);
}
#endif

// ---------------------------------------------------------------------------
// Init kernels
// ---------------------------------------------------------------------------
__global__ void build_w_kernel(float2* __restrict__ W) {
  int idx = blockIdx.x * blockDim.x + threadIdx.x;
  if (idx >= NFREQ * NFREQ) return;
  int u = idx >> 9, m = idx & 511;
  int t = (u * m) & 511;                    // phase mod 512 (exact in int)
  float th = -(float)(2.0 * M_PI / 512.0) * (float)t;
  float s, c;
  __sincosf(th, &s, &c);
  W[idx] = make_float2(c, s);
}

__global__ void build_r_kernel(float2* __restrict__ R, int P, int f) {
  int idx = blockIdx.x * blockDim.x + threadIdx.x;
  if (idx >= P * NFREQ) return;
  int i = idx >> 9, k = idx & 511;
  const float inv = 1.0f / 512.0f;
  float re, im;
  if (f == 1) {
    int t = (i * k) & 511;
    float th = (float)(2.0 * M_PI / 512.0) * (float)t;
    float s, c; __sincosf(th, &s, &c);
    re = c * inv; im = s * inv;
  } else {
    int a = f * i + (f >> 1) - 1;
    int b = a + 1;
    int ta = (a * k) & 511, tb = (b * k) & 511;
    float sa, ca, sb, cb;
    __sincosf((float)(2.0 * M_PI / 512.0) * (float)ta, &sa, &ca);
    __sincosf((float)(2.0 * M_PI / 512.0) * (float)tb, &sb, &cb);
    re = 0.5f * (ca + cb) * inv;
    im = 0.5f * (sa + sb) * inv;
  }
  R[idx] = make_float2(re, im);
}

__global__ void pack_input_kernel(const float* __restrict__ x,
                                  float2* __restrict__ xc, int n) {
  int i = blockIdx.x * blockDim.x + threadIdx.x;
  if (i < n) xc[i] = make_float2(x[i], 0.0f);
}

// ---------------------------------------------------------------------------
// Complex GEMM (NN): C[img][Mr x 512] = A[Mr x 512] @ (B[img][512x512] .* f1 .* f2)
// A tile staged by the TDM (interleaved complex, padded rows); filtered B tile
// staged on the vector path (filter fused at load). WG tile 64x64, wave 16x32.
// ---------------------------------------------------------------------------
__global__ __launch_bounds__(TPB)
void cgemm_nn_kernel(const float2* __restrict__ A,
                     const float2* __restrict__ B, long bImgStride,
                     const float* __restrict__ f1, const float* __restrict__ f2,
                     float2* __restrict__ C, long cImgStride,
                     int Mr) {
  __shared__ float sA[64 * TROW];
  __shared__ float sBr[16][65];
  __shared__ float sBi[16][65];

  const int tid   = threadIdx.x;
  const int img   = blockIdx.z;
  const int tileN = blockIdx.x * 64;
  const int tileM = blockIdx.y * 64;

  const float2* Bi = B + (long)img * bImgStride;
  float2*       Ci = C + (long)img * cImgStride;

  const int lane = tid & 31;
  const int wave = tid >> 5;
  const int wm   = wave & 3;
  const int wn   = wave >> 2;
  const int half = lane >> 4;
  const int l15  = lane & 15;

  v8f p_rr0 = {0,0,0,0,0,0,0,0}, p_ii0 = {0,0,0,0,0,0,0,0};
  v8f p_ri0 = {0,0,0,0,0,0,0,0}, p_ir0 = {0,0,0,0,0,0,0,0};
  v8f p_rr1 = {0,0,0,0,0,0,0,0}, p_ii1 = {0,0,0,0,0,0,0,0};
  v8f p_ri1 = {0,0,0,0,0,0,0,0}, p_ir1 = {0,0,0,0,0,0,0,0};

  for (int kb = 0; kb < NFREQ; kb += 16) {
    __syncthreads();
#if HAVE_TDM
    if (wave == 0) {
      // A tile: rows tileM..+63, complex cols kb..kb+15 -> 64 x 32 floats.
      tdm_load_tile_2d((unsigned)(uintptr_t)&sA[0],
                       A + (long)tileM * NFREQ + kb,
                       (unsigned)(2 * (NFREQ - kb)), (unsigned)(Mr - tileM),
                       32u, 64u, (unsigned)(2 * NFREQ));
    }
#else
#pragma unroll
    for (int i = 0; i < 4; ++i) {
      int idx = tid + i * TPB;
      int r   = idx >> 4;
      int kk  = idx & 15;
      int gr  = tileM + r; if (gr >= Mr) gr = Mr - 1;
      float2 v = A[(long)gr * NFREQ + kb + kk];
      sA[r * TROW + 2 * kk]     = v.x;
      sA[r * TROW + 2 * kk + 1] = v.y;
    }
#endif
    // Filtered B tile: 16 K x 64 cols (spectral filter fused -> vector path).
#pragma unroll
    for (int i = 0; i < 4; ++i) {
      int idx = tid + i * TPB;
      int r   = idx >> 6;
      int c   = idx & 63;
      long gi = (long)(kb + r) * NFREQ + tileN + c;
      float2 v = Bi[gi];
      float s = 1.0f;
      if (f1) s *= f1[gi];
      if (f2) s *= f2[gi];
      if (kb + 16 < NFREQ) __builtin_prefetch(&Bi[gi + 16L * NFREQ], 0, 0);
      sBr[r][c] = v.x * s;
      sBi[r][c] = v.y * s;
    }
#if HAVE_TDM
    if (wave == 0) __builtin_amdgcn_s_wait_tensorcnt(0);
#endif
    __syncthreads();

#pragma unroll
    for (int k0 = 0; k0 < 16; k0 += 4) {
      int kk = k0 + (half << 1);
      int ab = ((wm << 4) + l15) * TROW + 2 * kk;
      v2f aR, aI;
      aR.x = sA[ab];     aI.x = sA[ab + 1];
      aR.y = sA[ab + 2]; aI.y = sA[ab + 3];
      int bc = (wn << 5) + l15;
      v2f bR0, bI0, bR1, bI1;
      bR0.x = sBr[kk][bc];      bR0.y = sBr[kk + 1][bc];
      bI0.x = sBi[kk][bc];      bI0.y = sBi[kk + 1][bc];
      bR1.x = sBr[kk][bc + 16]; bR1.y = sBr[kk + 1][bc + 16];
      bI1.x = sBi[kk][bc + 16]; bI1.y = sBi[kk + 1][bc + 16];

      p_rr0 = WMMA_F32_16x16x4(aR, bR0, p_rr0);
      p_ii0 = WMMA_F32_16x16x4(aI, bI0, p_ii0);
      p_ri0 = WMMA_F32_16x16x4(aR, bI0, p_ri0);
      p_ir0 = WMMA_F32_16x16x4(aI, bR0, p_ir0);
      p_rr1 = WMMA_F32_16x16x4(aR, bR1, p_rr1);
      p_ii1 = WMMA_F32_16x16x4(aI, bI1, p_ii1);
      p_ri1 = WMMA_F32_16x16x4(aR, bI1, p_ri1);
      p_ir1 = WMMA_F32_16x16x4(aI, bR1, p_ir1);
    }
  }

  // Cr = ArBr - AiBi, Ci = ArBi + AiBr (f32 WMMA has C-neg only, no A/B NEG).
  const int rowBase = tileM + (wm << 4) + (half << 3);
  const int col0    = tileN + (wn << 5) + l15;
#pragma unroll
  for (int r = 0; r < 8; ++r) {
    int row = rowBase + r;
    if (row < Mr) {
      Ci[(long)row * NFREQ + col0]      = make_float2(p_rr0[r] - p_ii0[r], p_ri0[r] + p_ir0[r]);
      Ci[(long)row * NFREQ + col0 + 16] = make_float2(p_rr1[r] - p_ii1[r], p_ri1[r] + p_ir1[r]);
    }
  }
}

// ---------------------------------------------------------------------------
// Complex GEMM (NT): C[img][Mr x Nr] = A[img][Mr x 512] @ B[Nr x 512]^T
// Both operand tiles staged by the TDM (no filter on this pass). The B^T tile
// is just a 2D tile whose LDS row index is the output column.
// ---------------------------------------------------------------------------
__global__ __launch_bounds__(TPB)
void cgemm_nt_kernel(const float2* __restrict__ A, long aImgStride,
                     const float2* __restrict__ Bt,
                     float2* __restrict__ C, long cImgStride,
                     int Mr, int Nr) {
  __shared__ float sA[64 * TROW];
  __shared__ float sB[64 * TROW];

  const int tid   = threadIdx.x;
  const int img   = blockIdx.z;
  const int tileN = blockIdx.x * 64;
  const int tileM = blockIdx.y * 64;

  const float2* Ai = A + (long)img * aImgStride;
  float2*       Ci = C + (long)img * cImgStride;

  const int lane = tid & 31;
  const int wave = tid >> 5;
  const int wm   = wave & 3;
  const int wn   = wave >> 2;
  const int half = lane >> 4;
  const int l15  = lane & 15;

  v8f p_rr0 = {0,0,0,0,0,0,0,0}, p_ii0 = {0,0,0,0,0,0,0,0};
  v8f p_ri0 = {0,0,0,0,0,0,0,0}, p_ir0 = {0,0,0,0,0,0,0,0};
  v8f p_rr1 = {0,0,0,0,0,0,0,0}, p_ii1 = {0,0,0,0,0,0,0,0};
  v8f p_ri1 = {0,0,0,0,0,0,0,0}, p_ir1 = {0,0,0,0,0,0,0,0};

  for (int kb = 0; kb < NFREQ; kb += 16) {
    __syncthreads();
#if HAVE_TDM
    if (wave == 0) {
      tdm_load_tile_2d((unsigned)(uintptr_t)&sA[0],
                       Ai + (long)tileM * NFREQ + kb,
                       (unsigned)(2 * (NFREQ - kb)), (unsigned)(Mr - tileM),
                       32u, 64u, (unsigned)(2 * NFREQ));
      tdm_load_tile_2d((unsigned)(uintptr_t)&sB[0],
                       Bt + (long)tileN * NFREQ + kb,
                       (unsigned)(2 * (NFREQ - kb)), (unsigned)(Nr - tileN),
                       32u, 64u, (unsigned)(2 * NFREQ));
      __builtin_amdgcn_s_wait_tensorcnt(0);
    }
#else
#pragma unroll
    for (int i = 0; i < 4; ++i) {
      int idx = tid + i * TPB;
      int r   = idx >> 4;
      int kk  = idx & 15;
      int gr  = tileM + r; if (gr >= Mr) gr = Mr - 1;
      float2 v = Ai[(long)gr * NFREQ + kb + kk];
      sA[r * TROW + 2 * kk]     = v.x;
      sA[r * TROW + 2 * kk + 1] = v.y;
    }
#pragma unroll
    for (int i = 0; i < 4; ++i) {
      int idx = tid + i * TPB;
      int c   = idx >> 4;
      int r   = idx & 15;
      int gc  = tileN + c; if (gc >= Nr) gc = Nr - 1;
      float2 v = Bt[(long)gc * NFREQ + kb + r];
      sB[c * TROW + 2 * r]     = v.x;
      sB[c * TROW + 2 * r + 1] = v.y;
    }
#endif
    __syncthreads();

#pragma unroll
    for (int k0 = 0; k0 < 16; k0 += 4) {
      int kk = k0 + (half << 1);
      int ab = ((wm << 4) + l15) * TROW + 2 * kk;
      v2f aR, aI;
      aR.x = sA[ab];     aI.x = sA[ab + 1];
      aR.y = sA[ab + 2]; aI.y = sA[ab + 3];
      int bc = (wn << 5) + l15;
      int bb0 = bc * TROW + 2 * kk;
      int bb1 = (bc + 16) * TROW + 2 * kk;
      v2f bR0, bI0, bR1, bI1;
      bR0.x = sB[bb0];     bI0.x = sB[bb0 + 1];
      bR0.y = sB[bb0 + 2]; bI0.y = sB[bb0 + 3];
      bR1.x = sB[bb1];     bI1.x = sB[bb1 + 1];
      bR1.y = sB[bb1 + 2]; bI1.y = sB[bb1 + 3];

      p_rr0 = WMMA_F32_16x16x4(aR, bR0, p_rr0);
      p_ii0 = WMMA_F32_16x16x4(aI, bI0, p_ii0);
      p_ri0 = WMMA_F32_16x16x4(aR, bI0, p_ri0);
      p_ir0 = WMMA_F32_16x16x4(aI, bR0, p_ir0);
      p_rr1 = WMMA_F32_16x16x4(aR, bR1, p_rr1);
      p_ii1 = WMMA_F32_16x16x4(aI, bI1, p_ii1);
      p_ri1 = WMMA_F32_16x16x4(aR, bI1, p_ri1);
      p_ir1 = WMMA_F32_16x16x4(aI, bR1, p_ir1);
    }
  }

  const int rowBase = tileM + (wm << 4) + (half << 3);
  const int col0    = tileN + (wn << 5) + l15;
#pragma unroll
  for (int r = 0; r < 8; ++r) {
    int row = rowBase + r;
    if (row < Mr) {
      if (col0 < Nr)
        Ci[(long)row * Nr + col0] = make_float2(p_rr0[r] - p_ii0[r], p_ri0[r] + p_ir0[r]);
      if (col0 + 16 < Nr)
        Ci[(long)row * Nr + col0 + 16] = make_float2(p_rr1[r] - p_ii1[r], p_ri1[r] + p_ir1[r]);
    }
  }
}

// ---------------------------------------------------------------------------
// Host orchestration
// ---------------------------------------------------------------------------
extern "C" void kernel_launch(void* const* d_in, const int* in_sizes, int n_in,
                              void* d_out, int out_size, void* d_ws, size_t ws_size,
                              hipStream_t stream) {
  (void)in_sizes; (void)n_in; (void)out_size; (void)ws_size;

  const float* x   = (const float*)d_in[0];   // [4,3,512,512]
  const float* hp  = (const float*)d_in[1];   // [512,512]
  const float* lp  = (const float*)d_in[2];   // [512,512]
  const float* ang = (const float*)d_in[3];   // [4,512,512]
  const float* sc  = (const float*)d_in[4];   // [5,512,512]

  const long PLANE = (long)NFREQ * NFREQ;
  const long IMGP  = (long)NIMG * PLANE;

  float2* W  = (float2*)d_ws;
  float2* R  = W + PLANE;
  long rOff[6]; int rP[6];
  {
    long acc = 0;
    for (int p = 0; p < 6; ++p) { rP[p] = NFREQ >> p; rOff[p] = acc; acc += (long)rP[p] * NFREQ; }
  }
  const long RTOT = rOff[5] + (long)rP[5] * NFREQ;
  float2* xc = R + RTOT;
  float2* T  = xc + IMGP;
  float2* Xf = T + IMGP;
  float2* Z  = Xf + IMGP;

  float2* out = (float2*)d_out;
  float2* outHP = out;
  long off = IMGP;
  float2* outS[5];
  int   Ps[5];
  for (int j = 0; j < 5; ++j) {
    int P = NFREQ >> (j + 1);
    Ps[j] = P;
    outS[j] = out + off;
    off += (long)NIMG * 4 * P * P;
  }
  float2* outLP = out + off;

  {
    int n = (int)PLANE;
    build_w_kernel<<<(n + TPB - 1) / TPB, TPB, 0, stream>>>(W);
    for (int p = 0; p < 6; ++p) {
      int np = rP[p] * NFREQ;
      build_r_kernel<<<(np + TPB - 1) / TPB, TPB, 0, stream>>>(R + rOff[p], rP[p], 1 << p);
    }
    int nx = (int)IMGP;
    pack_input_kernel<<<(nx + TPB - 1) / TPB, TPB, 0, stream>>>(x, xc, nx);
  }

  // forward FFT: Xf = W @ x @ W^T
  {
    dim3 g(NFREQ / 64, NFREQ / 64, NIMG);
    cgemm_nn_kernel<<<g, TPB, 0, stream>>>(W, xc, PLANE, nullptr, nullptr, T, PLANE, NFREQ);
    cgemm_nt_kernel<<<g, TPB, 0, stream>>>(T, PLANE, W, Xf, PLANE, NFREQ, NFREQ);
  }

  // highpass residual (full res)
  {
    dim3 g(NFREQ / 64, NFREQ / 64, NIMG);
    cgemm_nn_kernel<<<g, TPB, 0, stream>>>(R + rOff[0], Xf, PLANE, hp, nullptr, Z, PLANE, NFREQ);
    cgemm_nt_kernel<<<g, TPB, 0, stream>>>(Z, PLANE, R + rOff[0], outHP, PLANE, NFREQ, NFREQ);
  }

  // oriented bands: R_f @ (Xf .* ang_k .* sc_j) @ R_f^T, f = 2^(j+1)
  for (int j = 0; j < 5; ++j) {
    int p = j + 1;
    int P = Ps[j];
    const float2* Rf = R + rOff[p];
    dim3 gnn(NFREQ / 64, (P + 63) / 64, NIMG);
    dim3 gnt((P + 63) / 64, (P + 63) / 64, NIMG);
    for (int k = 0; k < 4; ++k) {
      cgemm_nn_kernel<<<gnn, TPB, 0, stream>>>(Rf, Xf, PLANE,
                                               ang + (long)k * PLANE, sc + (long)j * PLANE,
                                               Z, (long)P * NFREQ, P);
      cgemm_nt_kernel<<<gnt, TPB, 0, stream>>>(Z, (long)P * NFREQ, Rf,
                                               outS[j] + (long)k * P * P, (long)4 * P * P,
                                               P, P);
    }
  }

  // lowpass
  {
    int P = 16;
    const float2* Rf = R + rOff[5];
    dim3 gnn(NFREQ / 64, 1, NIMG);
    dim3 gnt(1, 1, NIMG);
    cgemm_nn_kernel<<<gnn, TPB, 0, stream>>>(Rf, Xf, PLANE, lp, nullptr, Z, (long)P * NFREQ, P);
    cgemm_nt_kernel<<<gnt, TPB, 0, stream>>>(Z, (long)P * NFREQ, Rf, outLP, (long)P * P, P, P);
  }
}